// JambaMambaDecoderLayer_85787676770768
// MI455X (gfx1250) — compile-verified
//
#include <hip/hip_runtime.h>
#include <hip/hip_bf16.h>
#include <math.h>

typedef __attribute__((ext_vector_type(2))) float v2f;
typedef __attribute__((ext_vector_type(8))) float v8f;

#define NB 2
#define SEQ 2048
#define TOK 4096          // NB*SEQ
#define DM 1024
#define DI 2048
#define DTR 64
#define DST 16
#define DFF 4096

// Async global->LDS path (CDNA5 GLOBAL_LOAD_ASYNC_TO_LDS_*, ASYNCcnt) if the
// toolchain declares the builtins; otherwise fall back to sync loads.
#if defined(__has_builtin)
#if __has_builtin(__builtin_amdgcn_global_load_async_to_lds_b128) && \
    __has_builtin(__builtin_amdgcn_s_wait_asynccnt)
#define USE_ASYNC_LDS 1
#endif
#endif
#ifndef USE_ASYNC_LDS
#define USE_ASYNC_LDS 0
#endif

// Builtin param types (from hipcc diagnostic): pointer to 16B int vector,
// global (AS1) source / LDS (AS3) destination.
typedef int v4i_b128 __attribute__((vector_size(16)));
typedef __attribute__((address_space(1))) v4i_b128 glob_b128;
typedef __attribute__((address_space(3))) v4i_b128 lds_b128;

// ---------------------------------------------------------------------------
// FAST WMMA f32 GEMM: C[M,N] = A[M,K] @ B[K,N]
// Requires: M % 64 == 0, N % 64 == 0, K % 16 == 0 (no guards -> full EXEC).
// Block = 256 threads = 8 waves (4x2). Block tile 64x64, K-slice 16.
// Each wave: 16x32 (two v8f accumulators sharing one A fragment).
// ---------------------------------------------------------------------------
#define FBM 64
#define FBN 64
#define FKS 16
#define LDA_P 20   // padded row stride (floats); 80B keeps 16B alignment
#define LDB_P 68   // 272B, 16B-aligned

__global__ __launch_bounds__(256) void k_gemm_f32_fast(const float* __restrict__ A,
                                                       const float* __restrict__ B,
                                                       float* __restrict__ C,
                                                       int M, int N, int K) {
    __shared__ float As[FBM * LDA_P];
    __shared__ float Bs[FKS * LDB_P];

    const int tid  = threadIdx.x;
    const int wave = tid >> 5;
    const int lane = tid & 31;
    const int half = lane >> 4;     // 0: K 0,1 / M 0..7   1: K 2,3 / M 8..15
    const int r    = lane & 15;
    const int wm   = wave >> 1;     // 0..3 -> M offset wm*16
    const int wn   = wave & 1;      // 0..1 -> N offset wn*32

    const int blockM = blockIdx.y * FBM;
    const int blockN = blockIdx.x * FBN;

    // One B128 (4 floats) per thread per tile:
    const int arow = tid >> 2, acg = (tid & 3) * 4;    // A: 64x16
    const int brow = tid >> 4, bcg = (tid & 15) * 4;   // B: 16x64

    v8f acc0 = {}, acc1 = {};

    for (int k0 = 0; k0 < K; k0 += FKS) {
        const float* gA = &A[(size_t)(blockM + arow) * K + k0 + acg];
        const float* gB = &B[(size_t)(k0 + brow) * N + blockN + bcg];
        float* lA = &As[arow * LDA_P + acg];
        float* lB = &Bs[brow * LDB_P + bcg];
#if USE_ASYNC_LDS
        __builtin_amdgcn_global_load_async_to_lds_b128((glob_b128*)gA, (lds_b128*)lA, 0, 0);
        __builtin_amdgcn_global_load_async_to_lds_b128((glob_b128*)gB, (lds_b128*)lB, 0, 0);
#else
        float4 va = *(const float4*)gA;
        float4 vb = *(const float4*)gB;
        *(float4*)lA = va;
        *(float4*)lB = vb;
#endif
        if (k0 + FKS < K) {   // global_prefetch_b8 for next K-slice
            __builtin_prefetch(gA + FKS, 0, 1);
            __builtin_prefetch(gB + (size_t)FKS * N, 0, 1);
        }
#if USE_ASYNC_LDS
        __builtin_amdgcn_s_wait_asynccnt(0);
#endif
        __syncthreads();

#pragma unroll
        for (int k2 = 0; k2 < FKS; k2 += 4) {
            const int kk = k2 + 2 * half;
            v2f av, b0, b1;
            av.x = As[(wm * 16 + r) * LDA_P + kk];
            av.y = As[(wm * 16 + r) * LDA_P + kk + 1];
            b0.x = Bs[kk * LDB_P + wn * 32 + r];
            b0.y = Bs[(kk + 1) * LDB_P + wn * 32 + r];
            b1.x = Bs[kk * LDB_P + wn * 32 + 16 + r];
            b1.y = Bs[(kk + 1) * LDB_P + wn * 32 + 16 + r];
            acc0 = __builtin_amdgcn_wmma_f32_16x16x4_f32(
                false, av, false, b0, (short)0, acc0, false, false);
            acc1 = __builtin_amdgcn_wmma_f32_16x16x4_f32(
                false, av, false, b1, (short)0, acc1, false, false);
        }
        __syncthreads();
    }

    const int gmBase = blockM + wm * 16 + 8 * half;
    const int gn0 = blockN + wn * 32 + r;
#pragma unroll
    for (int i = 0; i < 8; ++i) {
        C[(size_t)(gmBase + i) * N + gn0]      = acc0[i];
        C[(size_t)(gmBase + i) * N + gn0 + 16] = acc1[i];
    }
}

// ---------------------------------------------------------------------------
// Guarded WMMA GEMM (for N not a multiple of 64, e.g. x_proj N=96).
// Block tile 32x64, 8 waves (2x4), one 16x16 tile per wave.
// ---------------------------------------------------------------------------
#define BM 32
#define BN 64
#define KS 16

__global__ __launch_bounds__(256) void k_gemm_f32_guard(const float* __restrict__ A,
                                                        const float* __restrict__ B,
                                                        float* __restrict__ C,
                                                        int M, int N, int K) {
    __shared__ float As[BM][KS + 1];
    __shared__ float Bs[KS][BN + 1];

    const int tid  = threadIdx.x;
    const int wave = tid >> 5;
    const int lane = tid & 31;
    const int half = lane >> 4;
    const int r    = lane & 15;
    const int wm = wave >> 2;
    const int wn = wave & 3;
    const int blockM = blockIdx.y * BM;
    const int blockN = blockIdx.x * BN;

    v8f acc = {};

    for (int k0 = 0; k0 < K; k0 += KS) {
        {
            int i = tid;
#pragma unroll
            for (int rep = 0; rep < 2; ++rep, i += 256) {
                int arow = i >> 4, acol = i & 15;
                As[arow][acol] = A[(size_t)(blockM + arow) * K + k0 + acol];
            }
        }
        {
            int i = tid;
#pragma unroll
            for (int rep = 0; rep < 4; ++rep, i += 256) {
                int brow = i >> 6, bcol = i & 63;
                int gn = blockN + bcol;
                Bs[brow][bcol] = (gn < N) ? B[(size_t)(k0 + brow) * N + gn] : 0.0f;
            }
        }
        __syncthreads();
#pragma unroll
        for (int k2 = 0; k2 < KS; k2 += 4) {
            v2f av, bv;
            av.x = As[wm * 16 + r][k2 + 2 * half];
            av.y = As[wm * 16 + r][k2 + 2 * half + 1];
            bv.x = Bs[k2 + 2 * half][wn * 16 + r];
            bv.y = Bs[k2 + 2 * half + 1][wn * 16 + r];
            acc = __builtin_amdgcn_wmma_f32_16x16x4_f32(
                false, av, false, bv, (short)0, acc, false, false);
        }
        __syncthreads();
    }
#pragma unroll
    for (int i = 0; i < 8; ++i) {
        int gm = blockM + wm * 16 + i + 8 * half;
        int gn = blockN + wn * 16 + r;
        if (gm < M && gn < N) C[(size_t)gm * N + gn] = acc[i];
    }
}

// ---------------------------------------------------------------------------
// RMSNorm over rows of length D (one block per row)
// ---------------------------------------------------------------------------
__global__ __launch_bounds__(256) void k_rmsnorm(const float* __restrict__ in,
                                                 const float* __restrict__ w,
                                                 float* __restrict__ out, int D) {
    const int row = blockIdx.x;
    const float* x = in + (size_t)row * D;
    float ss = 0.f;
    for (int i = threadIdx.x; i < D; i += 256) { float v = x[i]; ss += v * v; }
    __shared__ float red[256];
    red[threadIdx.x] = ss;
    __syncthreads();
    for (int s = 128; s > 0; s >>= 1) {
        if (threadIdx.x < s) red[threadIdx.x] += red[threadIdx.x + s];
        __syncthreads();
    }
    float inv = rsqrtf(red[0] / (float)D + 1e-6f);
    for (int i = threadIdx.x; i < D; i += 256)
        out[(size_t)row * D + i] = x[i] * inv * w[i];
}

// ---------------------------------------------------------------------------
// Per-token RMS of dt(64) / B(16) / C(16) slices from proj[T,96]
// ---------------------------------------------------------------------------
__global__ void k_norm_dtbc(const float* __restrict__ proj,
                            const float* __restrict__ dtw,
                            const float* __restrict__ bw,
                            const float* __restrict__ cw,
                            float* __restrict__ dtn, float* __restrict__ Bn,
                            float* __restrict__ Cn) {
    int t = blockIdx.x * blockDim.x + threadIdx.x;
    if (t >= TOK) return;
    const float* p = proj + (size_t)t * 96;
    float ss = 0.f;
    for (int i = 0; i < 64; ++i) { float v = p[i]; ss += v * v; }
    float inv = rsqrtf(ss / 64.f + 1e-6f);
    for (int i = 0; i < 64; ++i) dtn[(size_t)t * 64 + i] = p[i] * inv * dtw[i];
    ss = 0.f;
    for (int i = 0; i < 16; ++i) { float v = p[64 + i]; ss += v * v; }
    inv = rsqrtf(ss / 16.f + 1e-6f);
    for (int i = 0; i < 16; ++i) Bn[(size_t)t * 16 + i] = p[64 + i] * inv * bw[i];
    ss = 0.f;
    for (int i = 0; i < 16; ++i) { float v = p[80 + i]; ss += v * v; }
    inv = rsqrtf(ss / 16.f + 1e-6f);
    for (int i = 0; i < 16; ++i) Cn[(size_t)t * 16 + i] = p[80 + i] * inv * cw[i];
}

// ---------------------------------------------------------------------------
// Depthwise causal conv (K=4) + SiLU. x is the first DI cols of xz[T,2*DI].
// ---------------------------------------------------------------------------
__global__ void k_conv_silu(const float* __restrict__ xz,
                            const float* __restrict__ cw,
                            const float* __restrict__ cb,
                            float* __restrict__ xc) {
    size_t idx = (size_t)blockIdx.x * blockDim.x + threadIdx.x;
    if (idx >= (size_t)TOK * DI) return;
    int d = (int)(idx & (DI - 1));
    size_t bt = idx >> 11;                    // token index (b*SEQ + t)
    int t = (int)(bt & (SEQ - 1));
    float acc = cb[d];
#pragma unroll
    for (int k = 0; k < 4; ++k) {
        int tt = t - 3 + k;
        if (tt >= 0) acc += xz[(bt - (size_t)(t - tt)) * (2 * DI) + d] * cw[d * 4 + k];
    }
    xc[idx] = acc / (1.f + __expf(-acc));
}

__global__ void k_conv_state(const float* __restrict__ xz, float* __restrict__ cs) {
    int idx = blockIdx.x * blockDim.x + threadIdx.x;   // B*DI*4
    if (idx >= NB * DI * 4) return;
    int k = idx & 3;
    int d = (idx >> 2) & (DI - 1);
    int b = idx >> 13;
    cs[idx] = xz[((size_t)b * SEQ + (SEQ - 4 + k)) * (2 * DI) + d];
}

// ---------------------------------------------------------------------------
// dt = softplus(dt_lin + bias)   (in place)
// ---------------------------------------------------------------------------
__global__ void k_softplus_bias(float* __restrict__ dt, const float* __restrict__ bias) {
    size_t i = (size_t)blockIdx.x * blockDim.x + threadIdx.x;
    if (i >= (size_t)TOK * DI) return;
    float v = dt[i] + bias[i & (DI - 1)];
    dt[i] = (v > 20.f) ? v : log1pf(__expf(v));
}

// ---------------------------------------------------------------------------
// Selective scan: 16 states/channel in registers, B_t/C_t in LDS per step.
// grid = NB * (DI/256) = 16 blocks of 256 threads.
// ---------------------------------------------------------------------------
__global__ __launch_bounds__(256) void k_scan(const float* __restrict__ xc,
                                              const float* __restrict__ dt,
                                              const float* __restrict__ Bn,
                                              const float* __restrict__ Cn,
                                              const float* __restrict__ A_log,
                                              float* __restrict__ ys,
                                              float* __restrict__ ssm_out) {
    const int b = blockIdx.x >> 3;
    const int d = (blockIdx.x & 7) * 256 + threadIdx.x;
    float hS[DST], Ar[DST];
#pragma unroll
    for (int n = 0; n < DST; ++n) {
        hS[n] = 0.f;
        Ar[n] = -__expf(A_log[(size_t)d * DST + n]);
    }
    __shared__ float sB[DST], sC[DST];
    for (int t = 0; t < SEQ; ++t) {
        size_t tok = (size_t)b * SEQ + t;
        if (threadIdx.x < DST)            sB[threadIdx.x] = Bn[tok * DST + threadIdx.x];
        else if (threadIdx.x < 2 * DST)   sC[threadIdx.x - DST] = Cn[tok * DST + threadIdx.x - DST];
        __syncthreads();
        float dtv = dt[tok * DI + d];
        float xv  = xc[tok * DI + d];
        float dx = dtv * xv, y = 0.f;
#pragma unroll
        for (int n = 0; n < DST; ++n) {
            hS[n] = hS[n] * __expf(dtv * Ar[n]) + dx * sB[n];
            y += hS[n] * sC[n];
        }
        ys[tok * DI + d] = y;
        __syncthreads();
    }
#pragma unroll
    for (int n = 0; n < DST; ++n)
        ssm_out[((size_t)b * DI + d) * DST + n] = hS[n];
}

// ---------------------------------------------------------------------------
// y = (y_scan + x*D_skip) * silu(z)   (in place on ys)
// ---------------------------------------------------------------------------
__global__ void k_combine(float* __restrict__ ys, const float* __restrict__ xc,
                          const float* __restrict__ xz, const float* __restrict__ Dsk) {
    size_t i = (size_t)blockIdx.x * blockDim.x + threadIdx.x;
    if (i >= (size_t)TOK * DI) return;
    int d = (int)(i & (DI - 1));
    size_t tok = i >> 11;
    float z = xz[tok * (2 * DI) + DI + d];
    ys[i] = (ys[i] + xc[i] * Dsk[d]) * (z / (1.f + __expf(-z)));
}

__global__ void k_add(const float* __restrict__ a, const float* __restrict__ b,
                      float* __restrict__ c, size_t n) {
    size_t i = (size_t)blockIdx.x * blockDim.x + threadIdx.x;
    if (i < n) c[i] = a[i] + b[i];
}

__global__ void k_silu_mul(float* __restrict__ g, const float* __restrict__ u, size_t n) {
    size_t i = (size_t)blockIdx.x * blockDim.x + threadIdx.x;
    if (i < n) {
        float v = g[i];
        g[i] = (v / (1.f + __expf(-v))) * u[i];
    }
}

// ---------------------------------------------------------------------------
extern "C" void kernel_launch(void* const* d_in, const int* in_sizes, int n_in,
                              void* d_out, int out_size, void* d_ws, size_t ws_size,
                              hipStream_t stream) {
    const float* hidden  = (const float*)d_in[0];
    const float* inW     = (const float*)d_in[1];
    const float* convW   = (const float*)d_in[2];
    const float* convB   = (const float*)d_in[3];
    const float* xprojW  = (const float*)d_in[4];
    const float* dtW     = (const float*)d_in[5];
    const float* dtB     = (const float*)d_in[6];
    const float* dtLn    = (const float*)d_in[7];
    const float* bLn     = (const float*)d_in[8];
    const float* cLn     = (const float*)d_in[9];
    const float* Alog    = (const float*)d_in[10];
    const float* Dskip   = (const float*)d_in[11];
    const float* outW    = (const float*)d_in[12];
    const float* inLn    = (const float*)d_in[13];
    const float* moeLn   = (const float*)d_in[14];
    const float* gateW   = (const float*)d_in[15];
    const float* upW     = (const float*)d_in[16];
    const float* downW   = (const float*)d_in[17];

    float* out    = (float*)d_out;                       // [TOK*DM]
    float* out_cs = out + (size_t)TOK * DM;              // conv_state [NB*DI*4]
    float* out_ss = out_cs + (size_t)NB * DI * 4;        // ssm_state  [NB*DI*DST]

    const size_t MW = 1u << 20;
    float* ws   = (float*)d_ws;
    float* xz   = ws;                    // 16M floats: in_proj out, later gate, later silu*up
    float* xc   = ws + 16 * MW;          //  8M: conv+silu x, later low half of up
    float* dtb  = ws + 24 * MW;          //  8M: dt, later high half of up
    float* ysb  = ws + 32 * MW;          //  8M: scan y, later hn (4M)
    float* hbuf = ws + 40 * MW;          //  4M: h_norm, later mamba_out, later ff_out
    float* hid2 = ws + 44 * MW;          //  4M: residual after mamba
    float* proj = ws + 48 * MW;          //  393216
    float* dtn  = proj + (size_t)TOK * 96;   // 262144
    float* Bn   = dtn + (size_t)TOK * DTR;   // 65536
    float* Cn   = Bn + (size_t)TOK * DST;    // 65536
    float* up   = xc;                    // 16M region [16M,32M) = xc ∪ dtb
    float* hn   = ysb;                   // 4M within ysb

    auto gemm = [&](const float* A, const float* B, float* C, int M, int N, int K) {
        if ((M % FBM == 0) && (N % FBN == 0)) {
            dim3 grid(N / FBN, M / FBM);
            k_gemm_f32_fast<<<grid, 256, 0, stream>>>(A, B, C, M, N, K);
        } else {
            dim3 grid((N + BN - 1) / BN, (M + BM - 1) / BM);
            k_gemm_f32_guard<<<grid, 256, 0, stream>>>(A, B, C, M, N, K);
        }
    };
    auto eltb = [](size_t n) { return (unsigned)((n + 255) / 256); };

    // 1) input RMSNorm
    k_rmsnorm<<<TOK, 256, 0, stream>>>(hidden, inLn, hbuf, DM);
    // 2) in_proj: [TOK,DM] @ [DM,2*DI] -> xz
    gemm(hbuf, inW, xz, TOK, 2 * DI, DM);
    // 3) causal depthwise conv + silu -> xc ; 4) conv_state
    k_conv_silu<<<eltb((size_t)TOK * DI), 256, 0, stream>>>(xz, convW, convB, xc);
    k_conv_state<<<eltb((size_t)NB * DI * 4), 256, 0, stream>>>(xz, out_cs);
    // 5) x_proj: [TOK,DI] @ [DI,96] -> proj (guarded path, N=96)
    gemm(xc, xprojW, proj, TOK, DTR + 2 * DST, DI);
    // 6) per-token RMS of dt/B/C
    k_norm_dtbc<<<eltb(TOK), 256, 0, stream>>>(proj, dtLn, bLn, cLn, dtn, Bn, Cn);
    // 7) dt_proj: [TOK,64] @ [64,DI] -> dtb ; 8) softplus+bias in place
    gemm(dtn, dtW, dtb, TOK, DI, DTR);
    k_softplus_bias<<<eltb((size_t)TOK * DI), 256, 0, stream>>>(dtb, dtB);
    // 9) selective scan -> ysb, ssm_state
    k_scan<<<NB * (DI / 256), 256, 0, stream>>>(xc, dtb, Bn, Cn, Alog, ysb, out_ss);
    // 10) y = (y + x*D)*silu(z)
    k_combine<<<eltb((size_t)TOK * DI), 256, 0, stream>>>(ysb, xc, xz, Dskip);
    // 11) out_proj: [TOK,DI] @ [DI,DM] -> hbuf (mamba_out)
    gemm(ysb, outW, hbuf, TOK, DM, DI);
    // 12) residual
    k_add<<<eltb((size_t)TOK * DM), 256, 0, stream>>>(hidden, hbuf, hid2, (size_t)TOK * DM);
    // 13) pre-MoE RMSNorm -> hn
    k_rmsnorm<<<TOK, 256, 0, stream>>>(hid2, moeLn, hn, DM);
    // 14) gate, 15) up
    gemm(hn, gateW, xz, TOK, DFF, DM);
    gemm(hn, upW, up, TOK, DFF, DM);
    // 16) silu(gate)*up in place
    k_silu_mul<<<eltb((size_t)TOK * DFF), 256, 0, stream>>>(xz, up, (size_t)TOK * DFF);
    // 17) down: [TOK,DFF] @ [DFF,DM] -> hbuf (ff_out)
    gemm(xz, downW, hbuf, TOK, DM, DFF);
    // 18) final residual -> d_out
    k_add<<<eltb((size_t)TOK * DM), 256, 0, stream>>>(hid2, hbuf, out, (size_t)TOK * DM);
}